// RGCN_72499047956819
// MI455X (gfx1250) — compile-verified
//
#include <hip/hip_runtime.h>
#include <hip/hip_bf16.h>

typedef float v2f __attribute__((ext_vector_type(2)));
typedef float v8f __attribute__((ext_vector_type(8)));

// Hardware fp32 atomic add at device scope (single L2-side RMW, no CAS loop,
// no return -> STOREcnt fire-and-forget). CDNA5 GLOBAL_ATOMIC_ADD_F32.
__device__ __forceinline__ void atomic_fadd_dev(float* p, float v) {
    asm volatile("global_atomic_add_f32 %0, %1, off scope:SCOPE_DEV"
                 :: "v"(p), "v"(v) : "memory");
}

// ---------------------------------------------------------------------------
// Utility fills
// ---------------------------------------------------------------------------
__global__ void fill_u32_kernel(unsigned* __restrict__ p, unsigned v, int n) {
    int i = blockIdx.x * blockDim.x + threadIdx.x;
    if (i < n) p[i] = v;
}

__global__ void zero_f4_kernel(float4* __restrict__ p, int n4) {
    int i = blockIdx.x * blockDim.x + threadIdx.x;
    if (i < n4) p[i] = make_float4(0.f, 0.f, 0.f, 0.f);
}

// ---------------------------------------------------------------------------
// Degree counting (self-loop handled by initializing counts to 1)
// src/dst are flat [R*E]; counts are [R][N]
// ---------------------------------------------------------------------------
__global__ void deg_count_kernel(const int* __restrict__ src, const int* __restrict__ dst,
                                 unsigned* __restrict__ cnt_out, unsigned* __restrict__ cnt_in,
                                 int E, int Nn, int RE) {
    int i = blockIdx.x * blockDim.x + threadIdx.x;
    if (i >= RE) return;
    int r = i / E;
    atomicAdd(&cnt_out[(size_t)r * Nn + src[i]], 1u);
    atomicAdd(&cnt_in [(size_t)r * Nn + dst[i]], 1u);
}

// In-place u32 count -> float rsqrt(count)
__global__ void to_rsqrt_kernel(unsigned* __restrict__ p, int n) {
    int i = blockIdx.x * blockDim.x + threadIdx.x;
    if (i >= n) return;
    float v = (float)p[i];
    ((float*)p)[i] = rsqrtf(v);
}

// ---------------------------------------------------------------------------
// GEMM: C[M x Nout] = (A * rs_row) @ B, fp32 WMMA 16x16x4.
//   A: [M x K] row-major, row m scaled by rs[m]
//   B: [K x Nout] row-major
// One wave computes one 16x16 tile; K-loop of 32 wmma ops (K=128).
// A-operand layout (ISA 7.12.2, 32-bit A 16x4): lane = (k>=2?16:0)+m, vgpr=k&1
// B-operand mirrored; C/D: vgpr v holds row (v + 8*half), col = lane&15.
// ---------------------------------------------------------------------------
__global__ void gemm_rs_wmma_kernel(const float* __restrict__ A, const float* __restrict__ rs,
                                    const float* __restrict__ B, float* __restrict__ C,
                                    int M, int K, int Nout) {
    const int lane = threadIdx.x & 31;
    const int wid  = threadIdx.x >> 5;
    const int ntn  = Nout >> 4;
    const int total = (M >> 4) * ntn;
    int t = blockIdx.x * (blockDim.x >> 5) + wid;
    if (t >= total) return;                 // wave-uniform exit: EXEC all-1s for WMMA

    const int mt = t / ntn, nt = t % ntn;
    const int lid = lane & 15, half = lane >> 4;
    const int m = (mt << 4) + lid;          // A row handled by this lane
    const int n = (nt << 4) + lid;          // B col handled by this lane
    const float scale = rs[m];

    const float* ap = A + (size_t)m * K + 2 * half;      // pair (k+2h, k+2h+1)
    const float* bp = B + (size_t)(2 * half) * Nout + n;

    v8f c = {};
#pragma unroll 4
    for (int k = 0; k < K; k += 4) {
        v2f a = *(const v2f*)(ap + k);
        a[0] *= scale;
        a[1] *= scale;
        v2f b;
        b[0] = bp[(size_t)k * Nout];
        b[1] = bp[(size_t)(k + 1) * Nout];
        c = __builtin_amdgcn_wmma_f32_16x16x4_f32(false, a, false, b,
                                                  (short)0, c, false, false);
    }

    float* cp = C + (size_t)((mt << 4) + half * 8) * Nout + n;
#pragma unroll
    for (int i = 0; i < 8; ++i) cp[(size_t)i * Nout] = c[i];
}

// ---------------------------------------------------------------------------
// Edge scatter: one wave per edge, lane covers VEC floats of the feature dim.
// agg[dst] += h[src]   via hardware fp32 global atomics (L2-resident).
// ---------------------------------------------------------------------------
template <int VEC>
__global__ void scatter_kernel(const float* __restrict__ h, const int* __restrict__ src,
                               const int* __restrict__ dst, float* __restrict__ agg,
                               int E, int F) {
    const int lane = threadIdx.x & 31;
    int w  = (blockIdx.x * blockDim.x + threadIdx.x) >> 5;
    int nw = (gridDim.x * blockDim.x) >> 5;
    for (int e = w; e < E; e += nw) {
        const int s = src[e], d = dst[e];
        const float* hp = h   + (size_t)s * F + lane * VEC;
        float*       ap = agg + (size_t)d * F + lane * VEC;
        if (VEC == 4) {
            float4 v = *(const float4*)hp;
            atomic_fadd_dev(ap + 0, v.x);
            atomic_fadd_dev(ap + 1, v.y);
            atomic_fadd_dev(ap + 2, v.z);
            atomic_fadd_dev(ap + 3, v.w);
        } else {
            float2 v = *(const float2*)hp;
            atomic_fadd_dev(ap + 0, v.x);
            atomic_fadd_dev(ap + 1, v.y);
        }
    }
}

// ---------------------------------------------------------------------------
// Combine: acc += (agg + h) * rs_in[n] + bias[f]
// (h added here = self-loop contribution, avoiding N*F atomics)
// ---------------------------------------------------------------------------
__global__ void combine_kernel(float* __restrict__ acc, const float* __restrict__ agg,
                               const float* __restrict__ h, const float* __restrict__ rsin,
                               const float* __restrict__ bias, int Nn, int F) {
    int i = blockIdx.x * blockDim.x + threadIdx.x;
    const int per = F >> 2;
    if (i >= Nn * per) return;
    const int n  = i / per;
    const int f4 = (i - n * per) << 2;
    const float r = rsin[n];
    const size_t base = (size_t)n * F + f4;
    float4 ag = *(const float4*)(agg + base);
    float4 hh = *(const float4*)(h + base);
    float4 bb = *(const float4*)(bias + f4);
    float4 a  = *(float4*)(acc + base);
    a.x += (ag.x + hh.x) * r + bb.x;
    a.y += (ag.y + hh.y) * r + bb.y;
    a.z += (ag.z + hh.z) * r + bb.z;
    a.w += (ag.w + hh.w) * r + bb.w;
    *(float4*)(acc + base) = a;
}

__global__ void relu_f4_kernel(float4* __restrict__ p, int n4) {
    int i = blockIdx.x * blockDim.x + threadIdx.x;
    if (i >= n4) return;
    float4 v = p[i];
    v.x = v.x > 0.f ? v.x : 0.f;
    v.y = v.y > 0.f ? v.y : 0.f;
    v.z = v.z > 0.f ? v.z : 0.f;
    v.w = v.w > 0.f ? v.w : 0.f;
    p[i] = v;
}

// ---------------------------------------------------------------------------
// Orchestration
// ---------------------------------------------------------------------------
extern "C" void kernel_launch(void* const* d_in, const int* in_sizes, int n_in,
                              void* d_out, int out_size, void* d_ws, size_t ws_size,
                              hipStream_t stream) {
    const float* x  = (const float*)d_in[0];
    const int*   src = (const int*)d_in[1];
    const int*   dst = (const int*)d_in[2];
    const float* W1 = (const float*)d_in[3];
    const float* b1 = (const float*)d_in[4];
    const float* W2 = (const float*)d_in[5];
    const float* b2 = (const float*)d_in[6];
    float* out = (float*)d_out;

    const int R   = 3;
    const int IN  = 128;
    const int Nn  = in_sizes[0] / IN;           // 50000
    const int E   = in_sizes[1] / R;            // 800000
    const int HID = in_sizes[3] / (R * IN);     // 128
    const int OUT = in_sizes[5] / (R * HID);    // 64

    // Workspace carve-out
    char* ws = (char*)d_ws;
    size_t offs = 0;
    auto carve = [&](size_t bytes) -> char* {
        char* p = ws + offs;
        offs = (offs + bytes + 255) & ~(size_t)255;
        return p;
    };
    unsigned* cnt_out = (unsigned*)carve((size_t)R * Nn * 4);
    unsigned* cnt_in  = (unsigned*)carve((size_t)R * Nn * 4);
    float* hbuf = (float*)carve((size_t)Nn * HID * 4);
    float* agg  = (float*)carve((size_t)Nn * HID * 4);
    float* acc1 = (float*)carve((size_t)Nn * HID * 4);
    float* rs_out = (float*)cnt_out;  // converted in place
    float* rs_in  = (float*)cnt_in;

    const int BT = 256;
    auto blocks = [](int n, int bt) { return (n + bt - 1) / bt; };

    // ---- degrees (shared by both layers): counts start at 1 (self-loop) ----
    fill_u32_kernel<<<blocks(R * Nn, BT), BT, 0, stream>>>(cnt_out, 1u, R * Nn);
    fill_u32_kernel<<<blocks(R * Nn, BT), BT, 0, stream>>>(cnt_in, 1u, R * Nn);
    deg_count_kernel<<<blocks(R * E, BT), BT, 0, stream>>>(src, dst, cnt_out, cnt_in, E, Nn, R * E);
    to_rsqrt_kernel<<<blocks(R * Nn, BT), BT, 0, stream>>>(cnt_out, R * Nn);
    to_rsqrt_kernel<<<blocks(R * Nn, BT), BT, 0, stream>>>(cnt_in, R * Nn);

    // ---- layer 1: acc1 = sum_r GraphConv_r(x; W1_r, b1_r) ----
    zero_f4_kernel<<<blocks(Nn * HID / 4, BT), BT, 0, stream>>>((float4*)acc1, Nn * HID / 4);
    for (int r = 0; r < R; ++r) {
        const int tiles = (Nn / 16) * (HID / 16);
        gemm_rs_wmma_kernel<<<blocks(tiles, 8), BT, 0, stream>>>(
            x, rs_out + (size_t)r * Nn, W1 + (size_t)r * IN * HID, hbuf, Nn, IN, HID);
        zero_f4_kernel<<<blocks(Nn * HID / 4, BT), BT, 0, stream>>>((float4*)agg, Nn * HID / 4);
        scatter_kernel<4><<<8192, BT, 0, stream>>>(
            hbuf, src + (size_t)r * E, dst + (size_t)r * E, agg, E, HID);
        combine_kernel<<<blocks(Nn * HID / 4, BT), BT, 0, stream>>>(
            acc1, agg, hbuf, rs_in + (size_t)r * Nn, b1 + (size_t)r * HID, Nn, HID);
    }
    relu_f4_kernel<<<blocks(Nn * HID / 4, BT), BT, 0, stream>>>((float4*)acc1, Nn * HID / 4);

    // ---- layer 2: out = sum_r GraphConv_r(h1; W2_r, b2_r) ----
    zero_f4_kernel<<<blocks(Nn * OUT / 4, BT), BT, 0, stream>>>((float4*)out, Nn * OUT / 4);
    for (int r = 0; r < R; ++r) {
        const int tiles = (Nn / 16) * (OUT / 16);
        gemm_rs_wmma_kernel<<<blocks(tiles, 8), BT, 0, stream>>>(
            acc1, rs_out + (size_t)r * Nn, W2 + (size_t)r * HID * OUT, hbuf, Nn, HID, OUT);
        zero_f4_kernel<<<blocks(Nn * OUT / 4, BT), BT, 0, stream>>>((float4*)agg, Nn * OUT / 4);
        scatter_kernel<2><<<8192, BT, 0, stream>>>(
            hbuf, src + (size_t)r * E, dst + (size_t)r * E, agg, E, OUT);
        combine_kernel<<<blocks(Nn * OUT / 4, BT), BT, 0, stream>>>(
            out, agg, hbuf, rs_in + (size_t)r * Nn, b2 + (size_t)r * OUT, Nn, OUT);
    }
}